// MambaBlock_6897717477704
// MI455X (gfx1250) — compile-verified
//
#include <hip/hip_runtime.h>
#include <hip/hip_bf16.h>
#include <math.h>

// ---------------- problem constants (from reference) ----------------
#define D_MODEL   512
#define D_STATE   64
#define D_CONV    4
#define D_INNER   1024          // EXPAND * D_MODEL
#define DT_RANK   32            // ceil(512/16)
#define B_SZ      2
#define SEQ       1024
#define BL        (B_SZ * SEQ)  // 2048 token rows
#define XDBL_N    (DT_RANK + 2 * D_STATE)   // 160

typedef __attribute__((ext_vector_type(2))) float v2f;
typedef __attribute__((ext_vector_type(8))) float v8f;

// fp32 WMMA: D(16x16) = A(16x4) * B(4x16) + C  -- V_WMMA_F32_16X16X4_F32
static __device__ __forceinline__ v8f wmma_f32x4(v2f a, v2f b, v8f c) {
    return __builtin_amdgcn_wmma_f32_16x16x4_f32(false, a, false, b, (short)0, c,
                                                 false, false);
}

// CDNA5 async global->LDS copy (GLOBAL_LOAD_ASYNC_TO_LDS_B32, ASYNCcnt-tracked)
static __device__ __forceinline__ void async_copy_b32(unsigned lds_off,
                                                      const float* gptr) {
    asm volatile("global_load_async_to_lds_b32 %0, %1, off"
                 :
                 : "v"(lds_off), "v"(gptr)
                 : "memory");
}
static __device__ __forceinline__ void s_wait_asynccnt_16() {
    asm volatile("s_wait_asynccnt 0x10" ::: "memory");
}
static __device__ __forceinline__ void s_wait_asynccnt_0() {
    asm volatile("s_wait_asynccnt 0x0" ::: "memory");
}

static __device__ __forceinline__ float silu_f(float x) {
    return x / (1.0f + __expf(-x));
}
static __device__ __forceinline__ float softplus_f(float x) {
    return (x > 20.0f) ? x : log1pf(__expf(x));
}

// ---------------------------------------------------------------------------
// Stage one 16-wide K panel (A: 64x16 rows, B: 16x64 cols, B transposed in
// LDS) via async copies. Exactly 16 async instructions per thread, issued
// unconditionally (OOB B columns are address-clamped, not predicated), so
// every wave's ASYNCcnt advances by 16 per panel.
// ---------------------------------------------------------------------------
static __device__ __forceinline__
void stage_panel_async(const float* __restrict__ A, const float* __restrict__ B,
                       int lda, int ldb, int rowBase, int colBase, int N,
                       int k0, float (*As)[18], float (*BsT)[18], int tid) {
    #pragma unroll
    for (int i = tid; i < 64 * 16; i += 128) {          // 8 per thread
        const int r = i >> 4, c = i & 15;
        async_copy_b32((unsigned)(size_t)&As[r][c],
                       A + (size_t)(rowBase + r) * lda + (k0 + c));
    }
    #pragma unroll
    for (int i = tid; i < 16 * 64; i += 128) {          // 8 per thread
        const int r = i >> 6, c = i & 63;
        int gc = colBase + c;
        if (gc >= N) gc = N - 1;                        // clamp: uniform issue
        async_copy_b32((unsigned)(size_t)&BsT[c][r],
                       B + (size_t)(k0 + r) * ldb + gc);
    }
}

// ---------------------------------------------------------------------------
// Tiled fp32 WMMA GEMM: C[M x N] = A[M x K] @ B[K x N]  (row-major, strides)
//   128 threads (4 waves); block tile 64x64; K panel 16, double-buffered
//   async global->LDS staging. Requires M % 64 == 0, K % 16 == 0.
// EPI: 0 = none, 1 = bias + softplus
// ---------------------------------------------------------------------------
template <int EPI>
__global__ __launch_bounds__(128)
void gemm_wmma_f32(const float* __restrict__ A, const float* __restrict__ B,
                   float* __restrict__ C, const float* __restrict__ bias,
                   int M, int N, int K, int lda, int ldb, int ldc) {
    // pad 18 floats: rows 72 B (8-B aligned for b64 frag reads); gcd(18,64)=2
    // keeps 32 consecutive rows on distinct LDS banks.
    __shared__ __align__(16) float As[2][64][18];
    __shared__ __align__(16) float BsT[2][64][18];   // B stored [col][k]

    const int tid   = threadIdx.x;
    const int wave  = tid >> 5;
    const int lane  = tid & 31;
    const int lh    = lane >> 4;        // half-wave: K pair select / M+8
    const int l15   = lane & 15;

    const int rowBase = blockIdx.y * 64;
    const int colBase = blockIdx.x * 64;

    v8f acc[4];
    #pragma unroll
    for (int j = 0; j < 4; ++j) acc[j] = (v8f){0.f,0.f,0.f,0.f,0.f,0.f,0.f,0.f};

    const int P = K >> 4;               // number of 16-wide K panels
    stage_panel_async(A, B, lda, ldb, rowBase, colBase, N, 0,
                      As[0], BsT[0], tid);

    for (int p = 0; p < P; ++p) {
        const int buf = p & 1;
        if (p + 1 < P) {
            stage_panel_async(A, B, lda, ldb, rowBase, colBase, N,
                              (p + 1) << 4, As[buf ^ 1], BsT[buf ^ 1], tid);
            s_wait_asynccnt_16();       // retire current panel (in-order)
        } else {
            s_wait_asynccnt_0();
        }
        __syncthreads();                // all waves' panel data visible

        #pragma unroll
        for (int ks = 0; ks < 4; ++ks) {            // four K=4 sub-steps
            const int kk = ks * 4 + 2 * lh;         // even -> 8-B aligned
            const v2f a = *(const v2f*)&As[buf][wave * 16 + l15][kk];
            #pragma unroll
            for (int j = 0; j < 4; ++j) {
                const v2f b = *(const v2f*)&BsT[buf][j * 16 + l15][kk];
                acc[j] = wmma_f32x4(a, b, acc[j]);
            }
        }
        __syncthreads();                // buf free for refill in p+2
    }

    // epilogue: VGPR v -> lanes 0-15 = (M=v, N=l15), lanes 16-31 = (M=v+8)
    #pragma unroll
    for (int j = 0; j < 4; ++j) {
        const int col = colBase + j * 16 + l15;
        if (col < N) {
            #pragma unroll
            for (int v = 0; v < 8; ++v) {
                const int row = rowBase + wave * 16 + v + 8 * lh;
                float val = acc[j][v];
                if (EPI == 1) val = softplus_f(val + bias[col]);
                C[(size_t)row * ldc + col] = val;
            }
        }
    }
}

// ---------------------------------------------------------------------------
// Depthwise causal conv (width 4) + SiLU over xi = x_and_res[..., 0:1024]
// ---------------------------------------------------------------------------
__global__ __launch_bounds__(256)
void conv_silu_kernel(const float* __restrict__ x_and_res,
                      const float* __restrict__ conv_w,
                      const float* __restrict__ conv_b,
                      float* __restrict__ xc) {
    const size_t idx = (size_t)blockIdx.x * blockDim.x + threadIdx.x;
    if (idx >= (size_t)BL * D_INNER) return;
    const int d  = (int)(idx % D_INNER);
    const int t  = (int)((idx / D_INNER) % SEQ);
    const int b  = (int)(idx / ((size_t)D_INNER * SEQ));

    float acc = conv_b[d];
    #pragma unroll
    for (int k = 0; k < D_CONV; ++k) {
        const int tt = t + k - (D_CONV - 1);
        if (tt >= 0) {
            acc += x_and_res[((size_t)(b * SEQ + tt)) * (2 * D_INNER) + d] *
                   conv_w[d * D_CONV + k];
        }
    }
    xc[idx] = silu_f(acc);
}

// ---------------------------------------------------------------------------
// Selective scan: one wave per (b, d) channel; 2 states per lane (n, n+32).
// y[b,t,d] = (sum_n s_n * C_n + xc*D) * silu(res)
// ---------------------------------------------------------------------------
__global__ __launch_bounds__(256)
void scan_kernel(const float* __restrict__ delta,   // [BL, D_INNER] softplus'd
                 const float* __restrict__ xc,      // [BL, D_INNER]
                 const float* __restrict__ x_dbl,   // [BL, 160]; B@32, C@96
                 const float* __restrict__ x_and_res, // res at [.., 1024+d]
                 const float* __restrict__ A_log,   // [D_INNER, 64]
                 const float* __restrict__ D_param, // [D_INNER]
                 float* __restrict__ y) {           // [BL, D_INNER]
    const int wg   = blockIdx.x * (blockDim.x >> 5) + (threadIdx.x >> 5);
    const int lane = threadIdx.x & 31;
    const int b    = wg / D_INNER;
    const int d    = wg % D_INNER;
    const int n0   = lane;
    const int n1   = lane + 32;

    const float A0 = -__expf(A_log[(size_t)d * D_STATE + n0]);
    const float A1 = -__expf(A_log[(size_t)d * D_STATE + n1]);
    const float Dp = D_param[d];

    float s0 = 0.0f, s1 = 0.0f;
    for (int t = 0; t < SEQ; ++t) {
        const size_t bt = (size_t)b * SEQ + t;
        const float dt = delta[bt * D_INNER + d];
        const float u  = xc[bt * D_INNER + d];
        const float* xd = x_dbl + bt * XDBL_N;
        const float B0 = xd[DT_RANK + n0];
        const float B1 = xd[DT_RANK + n1];
        const float C0 = xd[DT_RANK + D_STATE + n0];
        const float C1 = xd[DT_RANK + D_STATE + n1];

        const float du = dt * u;
        s0 = __expf(dt * A0) * s0 + du * B0;
        s1 = __expf(dt * A1) * s1 + du * B1;

        float part = s0 * C0 + s1 * C1;
        #pragma unroll
        for (int off = 16; off >= 1; off >>= 1)
            part += __shfl_xor(part, off, 32);

        if (lane == 0) {
            const float r = x_and_res[bt * (2 * D_INNER) + D_INNER + d];
            y[bt * D_INNER + d] = (part + u * Dp) * silu_f(r);
        }
    }
}

// ---------------------------------------------------------------------------
extern "C" void kernel_launch(void* const* d_in, const int* in_sizes, int n_in,
                              void* d_out, int out_size, void* d_ws, size_t ws_size,
                              hipStream_t stream) {
    const float* x         = (const float*)d_in[0]; // [2,1024,512]
    const float* in_proj_w = (const float*)d_in[1]; // [512,2048]
    const float* conv_w    = (const float*)d_in[2]; // [1024,4]
    const float* conv_b    = (const float*)d_in[3]; // [1024]
    const float* x_proj_w  = (const float*)d_in[4]; // [1024,160]
    const float* dt_proj_w = (const float*)d_in[5]; // [32,1024]
    const float* dt_proj_b = (const float*)d_in[6]; // [1024]
    const float* A_log     = (const float*)d_in[7]; // [1024,64]
    const float* D_param   = (const float*)d_in[8]; // [1024]
    const float* out_proj_w= (const float*)d_in[9]; // [1024,512]
    float* out = (float*)d_out;                     // [2,1024,512]

    // workspace carve-out (fp32): ~43.3 MB total
    float* ws     = (float*)d_ws;
    float* xr     = ws;                                 // [BL, 2048] x_and_res
    float* xc     = xr    + (size_t)BL * 2 * D_INNER;   // [BL, 1024]
    float* xdbl   = xc    + (size_t)BL * D_INNER;       // [BL, 160]
    float* delta  = xdbl  + (size_t)BL * XDBL_N;        // [BL, 1024]
    float* yg     = delta + (size_t)BL * D_INNER;       // [BL, 1024]

    const dim3 blk(128);

    // 1) x_and_res = X @ in_proj_w        (2048 x 512 x 2048)
    gemm_wmma_f32<0><<<dim3(2 * D_INNER / 64, BL / 64), blk, 0, stream>>>(
        x, in_proj_w, xr, nullptr, BL, 2 * D_INNER, D_MODEL,
        D_MODEL, 2 * D_INNER, 2 * D_INNER);

    // 2) xc = silu(causal_conv4(xi) + b)
    {
        const size_t n = (size_t)BL * D_INNER;
        conv_silu_kernel<<<(unsigned)((n + 255) / 256), 256, 0, stream>>>(
            xr, conv_w, conv_b, xc);
    }

    // 3) x_dbl = xc @ x_proj_w            (2048 x 1024 x 160)
    gemm_wmma_f32<0><<<dim3((XDBL_N + 63) / 64, BL / 64), blk, 0, stream>>>(
        xc, x_proj_w, xdbl, nullptr, BL, XDBL_N, D_INNER,
        D_INNER, XDBL_N, XDBL_N);

    // 4) delta = softplus(x_dbl[:, :32] @ dt_proj_w + b)   (2048 x 32 x 1024)
    gemm_wmma_f32<1><<<dim3(D_INNER / 64, BL / 64), blk, 0, stream>>>(
        xdbl, dt_proj_w, delta, dt_proj_b, BL, D_INNER, DT_RANK,
        XDBL_N, D_INNER, D_INNER);

    // 5) selective scan + skip + gating -> yg
    {
        const int waves = B_SZ * D_INNER;          // 2048 (b,d) channels
        scan_kernel<<<waves / 8, 256, 0, stream>>>(
            delta, xc, xdbl, xr, A_log, D_param, yg);
    }

    // 6) out = yg @ out_proj_w            (2048 x 1024 x 512)
    gemm_wmma_f32<0><<<dim3(D_MODEL / 64, BL / 64), blk, 0, stream>>>(
        yg, out_proj_w, out, nullptr, BL, D_MODEL, D_INNER,
        D_INNER, D_MODEL, D_MODEL);
}